// GCNLSTMCell_3599182594879
// MI455X (gfx1250) — compile-verified
//
#include <hip/hip_runtime.h>
#include <hip/hip_bf16.h>
#include <math.h>

// Problem constants (fixed by the reference)
constexpr int Bc = 4;
constexpr int Nc = 20000;
constexpr int Ic = 64;
constexpr int Hc = 64;
constexpr int Ec = 320000;
constexpr int Cin  = Ic + Hc;   // 128
constexpr int Cout = 4 * Hc;    // 256
constexpr int ROWS = Bc * Nc;   // 80000

typedef __attribute__((ext_vector_type(2))) float v2f;
typedef __attribute__((ext_vector_type(8))) float v8f;

// ---------------------------------------------------------------------------
// 1) zero degree buffer (must be re-zeroed every call: harness does not
//    re-poison between replays)
// ---------------------------------------------------------------------------
__global__ void k_zero_deg(float* __restrict__ deg) {
    int n = blockIdx.x * blockDim.x + threadIdx.x;
    if (n < Nc) deg[n] = 0.0f;
}

// ---------------------------------------------------------------------------
// 2) indegree over real edges (weight 1 each)
// ---------------------------------------------------------------------------
__global__ void k_degree(const int* __restrict__ edge, float* __restrict__ deg) {
    int e = blockIdx.x * blockDim.x + threadIdx.x;
    if (e < Ec) {
        int dst = edge[Ec + e];
        atomicAdd(&deg[dst], 1.0f);
    }
}

// ---------------------------------------------------------------------------
// 3) dinv = rsqrt(deg + 2)   (self-loop fill value 2.0, deg always > 0)
// ---------------------------------------------------------------------------
__global__ void k_dinv(const float* __restrict__ deg, float* __restrict__ dinv) {
    int n = blockIdx.x * blockDim.x + threadIdx.x;
    if (n < Nc) dinv[n] = rsqrtf(deg[n] + 2.0f);
}

// ---------------------------------------------------------------------------
// 4) WMMA GEMM: xw[row, :] = [input|h][row, :] @ W   (fp32, K=128, Nout=256)
//    One wave = one 16x16 tile of xw; 32 chained V_WMMA_F32_16X16X4_F32.
// ---------------------------------------------------------------------------
__global__ void k_gemm_wmma(const float* __restrict__ x,
                            const float* __restrict__ h,
                            const float* __restrict__ W,
                            float* __restrict__ xw) {
    const int lane    = threadIdx.x & 31;
    const int wave    = threadIdx.x >> 5;
    const int tile    = blockIdx.x * 8 + wave;          // 80000 tiles total
    const int rowTile = tile >> 4;                      // 5000 row tiles
    const int colTile = tile & 15;                      // 16 col tiles
    const int hi      = lane >> 4;                      // K-half selector
    const int m       = lane & 15;

    const int row  = rowTile * 16 + m;                  // < 80000
    const int bidx = row / Nc;
    const int nidx = row - bidx * Nc;
    const int col  = colTile * 16 + m;                  // output column (N=lane%16)

    const float* __restrict__ xrow = x + ((size_t)bidx * Nc + nidx) * Ic;
    const float* __restrict__ hrow = h + ((size_t)bidx * Nc + nidx) * Hc;

    v8f c = {};
#pragma unroll 4
    for (int k0 = 0; k0 < Cin; k0 += 4) {
        const float* srcrow = (k0 < Ic) ? xrow : hrow;
        const int coff = (k0 < Ic) ? k0 : (k0 - Ic);

        // A: 16x4 fp32 tile. lane<16 -> K={k0,k0+1}, lane>=16 -> K={k0+2,k0+3}
        v2f a = *(const v2f*)(srcrow + coff + 2 * hi);  // 8B aligned (even float off)

        // B: 4x16 fp32 tile of W, same K-half split, N = lane%16
        v2f bb;
        bb.x = W[(size_t)(k0 + 2 * hi) * Cout + col];
        bb.y = W[(size_t)(k0 + 2 * hi + 1) * Cout + col];

        // 8-arg form: (neg_a, A, neg_b, B, c_mod, C, reuse_a, reuse_b)
        c = __builtin_amdgcn_wmma_f32_16x16x4_f32(false, a, false, bb,
                                                  (short)0, c, false, false);
    }

    // D layout: VGPR j -> M = j + 8*hi, N = lane%16
    const int orow0 = rowTile * 16 + 8 * hi;
#pragma unroll
    for (int j = 0; j < 8; ++j) {
        xw[(size_t)(orow0 + j) * Cout + col] = c[j];
    }
}

// ---------------------------------------------------------------------------
// 5) conv init with self-loop term: conv[b,n,:] = (2*dinv[n]^2) * xw[b,n,:]
// ---------------------------------------------------------------------------
__global__ void k_selfloop(const float* __restrict__ xw,
                           const float* __restrict__ dinv,
                           float* __restrict__ conv) {
    size_t i = (size_t)blockIdx.x * blockDim.x + threadIdx.x;   // over ROWS*Cout
    if (i < (size_t)ROWS * Cout) {
        int row = (int)(i >> 8);              // /256
        int n   = row % Nc;
        float di = dinv[n];
        conv[i] = 2.0f * di * di * xw[i];
    }
}

// ---------------------------------------------------------------------------
// 6) edge scatter: conv[b,dst,:] += dinv[src]*dinv[dst] * xw[b,src,:]
//    One edge per wave; each lane handles 8 of 256 channels; atomics land in
//    L2 (xw+conv = 164MB fits the 192MB L2).
// ---------------------------------------------------------------------------
__global__ void k_scatter(const int* __restrict__ edge,
                          const float* __restrict__ dinv,
                          const float* __restrict__ xw,
                          float* __restrict__ conv) {
    const int lane = threadIdx.x & 31;
    const int wave = threadIdx.x >> 5;
    const int e    = blockIdx.x * 8 + wave;
    if (e >= Ec) return;

    const int src = edge[e];
    const int dst = edge[Ec + e];
    const float norm = dinv[src] * dinv[dst];
    const int ch = lane * 8;                          // 8 channels per lane

#pragma unroll
    for (int b = 0; b < Bc; ++b) {
        const float* __restrict__ sp = xw   + ((size_t)b * Nc + src) * Cout + ch;
        float* __restrict__       dp = conv + ((size_t)b * Nc + dst) * Cout + ch;
        if (b + 1 < Bc)
            __builtin_prefetch(xw + ((size_t)(b + 1) * Nc + src) * Cout + ch, 0, 0);
        float4 v0 = *(const float4*)(sp);
        float4 v1 = *(const float4*)(sp + 4);
        atomicAdd(dp + 0, norm * v0.x);
        atomicAdd(dp + 1, norm * v0.y);
        atomicAdd(dp + 2, norm * v0.z);
        atomicAdd(dp + 3, norm * v0.w);
        atomicAdd(dp + 4, norm * v1.x);
        atomicAdd(dp + 5, norm * v1.y);
        atomicAdd(dp + 6, norm * v1.z);
        atomicAdd(dp + 7, norm * v1.w);
    }
}

// ---------------------------------------------------------------------------
// 7) fused bias + LSTM gates: out = [h_next (B*N*H) || c_next (B*N*H)]
// ---------------------------------------------------------------------------
__device__ __forceinline__ float sigmoidf(float x) {
    return 1.0f / (1.0f + __expf(-x));
}

__global__ void k_gates(const float* __restrict__ conv,
                        const float* __restrict__ bias,
                        const float* __restrict__ c_cur,
                        float* __restrict__ out) {
    size_t i = (size_t)blockIdx.x * blockDim.x + threadIdx.x;   // over ROWS*H
    if (i >= (size_t)ROWS * Hc) return;
    int c   = (int)(i & (Hc - 1));
    int row = (int)(i >> 6);                                    // /64

    const float* cb = conv + (size_t)row * Cout;
    float cc_i = cb[c]            + bias[c];
    float cc_f = cb[Hc + c]       + bias[Hc + c];
    float cc_o = cb[2 * Hc + c]   + bias[2 * Hc + c];
    float cc_g = cb[3 * Hc + c]   + bias[3 * Hc + c];

    float ig = sigmoidf(cc_i);
    float fg = sigmoidf(cc_f);
    float og = sigmoidf(cc_o);
    float gg = tanhf(cc_g);

    float cn = fg * c_cur[i] + ig * gg;
    float hn = og * tanhf(cn);

    out[i]                      = hn;   // h_next
    out[(size_t)ROWS * Hc + i]  = cn;   // c_next
}

// ---------------------------------------------------------------------------
// launch
// ---------------------------------------------------------------------------
extern "C" void kernel_launch(void* const* d_in, const int* in_sizes, int n_in,
                              void* d_out, int out_size, void* d_ws, size_t ws_size,
                              hipStream_t stream) {
    const float* input  = (const float*)d_in[0];   // (B,N,I)
    const float* h_cur  = (const float*)d_in[1];   // (B,N,H)
    const float* c_cur  = (const float*)d_in[2];   // (B,N,H)
    const float* W      = (const float*)d_in[3];   // (128,256)
    const float* b_gcn  = (const float*)d_in[4];   // (256,)
    const int*   edge   = (const int*)d_in[5];     // (2,E)
    float*       out    = (float*)d_out;

    // workspace layout (floats)
    float* xw   = (float*)d_ws;                       // ROWS*Cout = 20,480,000
    float* conv = xw + (size_t)ROWS * Cout;           // ROWS*Cout
    float* deg  = conv + (size_t)ROWS * Cout;         // N
    float* dinv = deg + Nc;                           // N

    // 1) deg = 0
    k_zero_deg<<<(Nc + 255) / 256, 256, 0, stream>>>(deg);
    // 2) indegree
    k_degree<<<(Ec + 255) / 256, 256, 0, stream>>>(edge, deg);
    // 3) dinv
    k_dinv<<<(Nc + 255) / 256, 256, 0, stream>>>(deg, dinv);
    // 4) WMMA GEMM: 80000 tiles, 8 waves/block
    k_gemm_wmma<<<(ROWS / 16) * 16 / 8, 256, 0, stream>>>(input, h_cur, W, xw);
    // 5) self-loop init of conv
    {
        size_t tot = (size_t)ROWS * Cout;
        k_selfloop<<<(unsigned)((tot + 255) / 256), 256, 0, stream>>>(xw, dinv, conv);
    }
    // 6) edge scatter, one edge per wave
    k_scatter<<<(Ec + 7) / 8, 256, 0, stream>>>(edge, dinv, xw, conv);
    // 7) gates
    {
        size_t tot = (size_t)ROWS * Hc;
        k_gates<<<(unsigned)((tot + 255) / 256), 256, 0, stream>>>(conv, b_gcn, c_cur, out);
    }
}